// ScatLayerj1a_4440996184632
// MI455X (gfx1250) — compile-verified
//
#include <hip/hip_runtime.h>
#include <math.h>
#include <stdint.h>

typedef __attribute__((ext_vector_type(2))) float v2f;
typedef __attribute__((ext_vector_type(8))) float v8f;

#define N_B     16
#define C_CH    64
#define HIN     224
#define WIN     224
#define HOUT    112
#define WOUT    112
#define TPD     7            // tiles per spatial dim (112/16)
#define TILE    32           // full-res tile edge
#define KROWS   40           // 32 + 3 + 3 halo, padded to 40 (K dim, mult of 4)
#define XCOLS   38           // 32 + 3 + 3 columns needed
#define LSTRIDE 33           // padded LDS row stride for 32-wide arrays

__device__ __forceinline__ int reflect224(int i) {
    // half-sample symmetric reflection into [0, 224)
    i = (i < 0) ? (-1 - i) : i;
    i = (i >= HIN) ? (2 * HIN - 1 - i) : i;
    return i;
}

__device__ __forceinline__ int iclamp(int v, int lo, int hi) {
    return v < lo ? lo : (v > hi ? hi : v);
}

__global__ __launch_bounds__(256) void scat_fused_kernel(
        const float* __restrict__ x,
        const float* __restrict__ h0o,
        const float* __restrict__ h1o,
        float* __restrict__ out) {
    __shared__ float xin[KROWS][40];            // input tile with halo
    __shared__ float loh[KROWS][LSTRIDE];       // W-filtered (5-tap), with H halo
    __shared__ float hih[KROWS][LSTRIDE];       // W-filtered (7-tap), with H halo
    __shared__ float res[4][TILE][LSTRIDE];     // ll, lh, hl, hh full-res tiles
    __shared__ v2f   afr[2][2][10][32];         // A-fragment table: [filt][mhalf][kc][lane]
    __shared__ float h0s[5];
    __shared__ float h1s[7];

    const int tid  = threadIdx.x;
    const int blk  = blockIdx.x;
    const int plane = blk / (TPD * TPD);        // n*64 + c
    const int tIdx  = blk % (TPD * TPD);
    const int ty = tIdx / TPD;
    const int tx = tIdx % TPD;
    const int n   = plane >> 6;
    const int cch = plane & 63;

    const float* xp = x + (size_t)plane * (HIN * WIN);
    const int rowBase = ty * TILE - 3;
    const int colBase = tx * TILE - 3;

    // ---- Stage 0a: kick off async DMA of the input tile into LDS (ASYNCcnt) ----
    for (int idx = tid; idx < KROWS * XCOLS; idx += 256) {
        const int j = idx / XCOLS;
        const int c = idx % XCOLS;
        const int gr = reflect224(rowBase + j);
        const int gc = reflect224(colBase + c);
        const float* gp = xp + gr * WIN + gc;
        const uint32_t dst = (uint32_t)(uintptr_t)&xin[j][c];  // LDS byte address
        asm volatile("global_load_async_to_lds_b32 %0, %1, off"
                     :: "v"(dst), "v"(gp) : "memory");
    }

    // ---- Stage 0b (overlaps the DMA): filter taps + A-fragment table ----
    if (tid < 5) h0s[tid] = h0o[tid];
    if (tid < 7) h1s[tid] = h1o[tid];
    __syncthreads();   // publishes h0s/h1s; does NOT wait on ASYNCcnt

    // Band matrices: F0[r][k] = h0[k-r-1] (5 taps), F1[r][k] = h1[k-r] (7 taps)
    // A layout (16x4 f32): lane<16 holds K={0,1}, lane>=16 holds K={2,3}, M=lane&15
    for (int t = tid; t < 2 * 2 * 10 * 32; t += 256) {
        const int lane_ = t & 31;
        const int rest  = t >> 5;            // 0..39
        const int kc    = rest % 10;
        const int mh    = (rest / 10) & 1;
        const int filt  = rest / 20;         // 0 = h0 band, 1 = h1 band
        const int row   = mh * 16 + (lane_ & 15);
        const int kk    = kc * 4 + ((lane_ >> 4) << 1);
        const float* hs = filt ? h1s : h0s;
        const int ntap  = filt ? 7 : 5;
        const int doff  = filt ? 0 : 1;
        const int d0 = kk - row - doff;
        const int d1 = d0 + 1;
        float a0 = hs[iclamp(d0, 0, ntap - 1)];
        float a1 = hs[iclamp(d1, 0, ntap - 1)];
        a0 = (d0 >= 0 && d0 < ntap) ? a0 : 0.f;
        a1 = (d1 >= 0 && d1 < ntap) ? a1 : 0.f;
        afr[filt][mh][kc][lane_] = (v2f){a0, a1};
    }

    // Wait for this wave's async tile DMA, then make it visible block-wide.
    asm volatile("s_wait_asynccnt 0x0" ::: "memory");
    __syncthreads();

    // ---- Stage 1: W-direction filtering (VALU, cheap) ----
    // output col c corresponds to xin col c+3 (global col tx*32 + c)
    for (int idx = tid; idx < KROWS * TILE; idx += 256) {
        const int j = idx >> 5;
        const int c = idx & 31;
        const float* r = &xin[j][c];
        float lo = h0s[0] * r[1] + h0s[1] * r[2] + h0s[2] * r[3]
                 + h0s[3] * r[4] + h0s[4] * r[5];
        float hi = h1s[0] * r[0] + h1s[1] * r[1] + h1s[2] * r[2]
                 + h1s[3] * r[3] + h1s[4] * r[4] + h1s[5] * r[5]
                 + h1s[6] * r[6];
        loh[j][c] = lo;
        hih[j][c] = hi;
    }
    __syncthreads();

    // ---- Stage 2: H-direction filtering as banded matmul on the matrix pipe ----
    // Y(32x32) = F(32x40) x SRC(40x32); 16 subtiles over 8 waves, 2 each.
    const int wave  = tid >> 5;
    const int lane  = tid & 31;
    const int lrow  = lane & 15;          // M (for A) / N (for B) index
    const int khalf = (lane >> 4) << 1;   // 0 or 2: K sub-offset per f32 B layout

    #pragma unroll
    for (int si = 0; si < 2; ++si) {
        const int s   = wave * 2 + si;
        const int mat = s >> 2;                       // 0=ll 1=lh 2=hl 3=hh
        const int mh  = (s >> 1) & 1;
        const int m0  = mh * 16;
        const int n0  = (s & 1) * 16;
        const float (*src)[LSTRIDE] = (mat < 2) ? loh : hih;
        const int use7 = mat & 1;
        const v2f* at = &afr[use7][mh][0][lane];      // stride 32 per kc

        v8f acc = {0.f, 0.f, 0.f, 0.f, 0.f, 0.f, 0.f, 0.f};
        #pragma unroll
        for (int kc = 0; kc < KROWS / 4; ++kc) {
            const int kk = kc * 4 + khalf;
            v2f A = at[kc * 32];                                  // ds_load_b64
            v2f B = {src[kk][n0 + lrow], src[kk + 1][n0 + lrow]}; // 2x ds_load_b32
            acc = __builtin_amdgcn_wmma_f32_16x16x4_f32(
                      false, A, false, B, (short)0, acc, false, false);
        }

        // D layout: VGPR v -> row m0 + v (+8 for upper lane half), col n0 + lane&15
        const int drow = m0 + ((lane >> 4) ? 8 : 0);
        const int dcol = n0 + lrow;
        #pragma unroll
        for (int v = 0; v < 8; ++v)
            res[mat][drow + v][dcol] = acc[v];
    }
    __syncthreads();

    // ---- Stage 3: q2c, magnitude, maxpool, store 7 bands ----
    const int oy = tid >> 4;
    const int ox = tid & 15;
    const int Y0 = oy * 2, X0 = ox * 2;
    const int gY = ty * 16 + oy;
    const int gX = tx * 16 + ox;
    const float S = 0.7071067811865476f;

    // ll -> 2x2 maxpool
    float l00 = res[0][Y0][X0],     l01 = res[0][Y0][X0 + 1];
    float l10 = res[0][Y0 + 1][X0], l11 = res[0][Y0 + 1][X0 + 1];
    float llp = fmaxf(fmaxf(l00, l01), fmaxf(l10, l11));

    // lh -> d15, d165
    float a = res[1][Y0][X0],     b = res[1][Y0][X0 + 1];
    float c = res[1][Y0 + 1][X0], d = res[1][Y0 + 1][X0 + 1];
    float d15r = (a - d) * S, d15i = (b + c) * S;
    float d165r = (a + d) * S, d165i = (b - c) * S;
    // hh -> d45, d135
    a = res[3][Y0][X0];     b = res[3][Y0][X0 + 1];
    c = res[3][Y0 + 1][X0]; d = res[3][Y0 + 1][X0 + 1];
    float d45r = (a - d) * S, d45i = (b + c) * S;
    float d135r = (a + d) * S, d135i = (b - c) * S;
    // hl -> d75, d105
    a = res[2][Y0][X0];     b = res[2][Y0][X0 + 1];
    c = res[2][Y0 + 1][X0]; d = res[2][Y0 + 1][X0 + 1];
    float d75r = (a - d) * S, d75i = (b + c) * S;
    float d105r = (a + d) * S, d105i = (b - c) * S;

    const size_t bstride = (size_t)C_CH * HOUT * WOUT;
    const size_t obase = ((size_t)n * (7 * C_CH) + cch) * (HOUT * WOUT)
                       + (size_t)gY * WOUT + gX;

    out[obase] = llp;
    out[obase + 1 * bstride] = sqrtf(d15r * d15r + d15i * d15i + 1e-4f) - 0.01f;
    out[obase + 2 * bstride] = sqrtf(d45r * d45r + d45i * d45i + 1e-4f) - 0.01f;
    out[obase + 3 * bstride] = sqrtf(d75r * d75r + d75i * d75i + 1e-4f) - 0.01f;
    out[obase + 4 * bstride] = sqrtf(d105r * d105r + d105i * d105i + 1e-4f) - 0.01f;
    out[obase + 5 * bstride] = sqrtf(d135r * d135r + d135i * d135i + 1e-4f) - 0.01f;
    out[obase + 6 * bstride] = sqrtf(d165r * d165r + d165i * d165i + 1e-4f) - 0.01f;
}

extern "C" void kernel_launch(void* const* d_in, const int* in_sizes, int n_in,
                              void* d_out, int out_size, void* d_ws, size_t ws_size,
                              hipStream_t stream) {
    const float* x   = (const float*)d_in[0];
    const float* h0o = (const float*)d_in[1];
    const float* h1o = (const float*)d_in[2];
    float* out = (float*)d_out;

    const int nblocks = N_B * C_CH * TPD * TPD;  // 50176
    scat_fused_kernel<<<nblocks, 256, 0, stream>>>(x, h0o, h1o, out);
}